// Dynamic_sampling_M_86526411145606
// MI455X (gfx1250) — compile-verified
//
#include <hip/hip_runtime.h>
#include <hip/hip_bf16.h>

// ---------------------------------------------------------------------------
// Dynamic sampling / KNN-gather for MI455X (gfx1250, wave32).
//
// x (8, 5, 16384) f32, s_num = 4096 -> out (8, 5, 4096, 16) f32.
//
// Hot stage (16-NN in 3D) uses V_WMMA_F32_16X16X4_F32:
//   A (16 points x 4)  = [x, y, z, |p|^2]
//   B (4 x 16 queries) = [-2qx, -2qy, -2qz, 1] per column
//   D[m][n] = |p_m|^2 - 2 q_n . p_m     (query norm constant per query)
// Each lane owns 8 candidate distances for one query per WMMA tile; top-16
// kept as a branchless min/max insertion network over packed 64-bit keys
//   key = (monotone_uint(dist) << 32) | point_index
// which lowers to native v_min_u64/v_max_u64 on gfx1250 and gives exact
// lower-index tie-breaking (matching jax.lax.top_k).  Point channels are
// double-buffered into LDS with the CDNA5 async global->LDS path
// (global_load_async_to_lds_b128 + s_wait_asynccnt), overlapping global
// latency with a full chunk of WMMA work; the tile loop is software-
// pipelined so ds_load latency hides behind the insertion network.
// ---------------------------------------------------------------------------

typedef __attribute__((ext_vector_type(2))) float v2f;
typedef __attribute__((ext_vector_type(8))) float v8f;
typedef __attribute__((ext_vector_type(4))) int   v4i;

#define BATCHES 8
#define NPTS    16384
#define NSAMP   4096
#define KNN     16
#define CHUNK   1024
#define NCHUNK  (NPTS / CHUNK)
#define NTILE   (CHUNK / 16)
#define BIGF    3.0e38f

#if defined(__has_builtin)
#  if __has_builtin(__builtin_amdgcn_global_load_async_to_lds_b128)
#    define HAVE_ASYNC_LDS 1
#  endif
#endif
#ifndef HAVE_ASYNC_LDS
#  define HAVE_ASYNC_LDS 0
#endif

#if HAVE_ASYNC_LDS
#  if __has_builtin(__builtin_amdgcn_s_wait_asynccnt)
#    define WAIT_ASYNC() __builtin_amdgcn_s_wait_asynccnt(0)
#  else
#    define WAIT_ASYNC() asm volatile("s_wait_asynccnt 0x0" ::: "memory")
#  endif
// Builtin signature (from hipcc diagnostic): v4i addrspace(1)* src,
// v4i addrspace(3)* dst, imm int offset, imm int cpol.
typedef __attribute__((address_space(1))) v4i gv4i;
typedef __attribute__((address_space(3))) v4i lv4i;
#endif

// Monotone float -> uint map: a < b  <=>  ord(a) < ord(b)  (totally ordered).
__device__ __forceinline__ unsigned ord_f32(float f) {
    unsigned u = __float_as_uint(f);
    return u ^ ((unsigned)((int)u >> 31) | 0x80000000u);
}
__device__ __forceinline__ unsigned long long pack_key(float d, int idx) {
    return ((unsigned long long)ord_f32(d) << 32) | (unsigned)idx;
}
__device__ __forceinline__ unsigned long long umin64(unsigned long long a,
                                                     unsigned long long b) {
    return a < b ? a : b;
}
__device__ __forceinline__ unsigned long long umax64(unsigned long long a,
                                                     unsigned long long b) {
    return a > b ? a : b;
}

// ---------------------------------------------------------------------------
// Kernel 1: per (batch, channel 0/1) min & max over the 16384 points.
// ---------------------------------------------------------------------------
__global__ void minmax_kernel(const float* __restrict__ x,
                              float* __restrict__ minmax) {
    int bc = blockIdx.x;                 // b*2 + c, 16 blocks
    int b = bc >> 1, c = bc & 1;
    const float* xc = x + ((size_t)b * 5 + c) * NPTS;
    float mn = BIGF, mx = -BIGF;
    for (int t = threadIdx.x; t < NPTS; t += 256) {
        float v = xc[t];
        mn = fminf(mn, v);
        mx = fmaxf(mx, v);
    }
    __shared__ float smn[256], smx[256];
    smn[threadIdx.x] = mn;
    smx[threadIdx.x] = mx;
    __syncthreads();
    for (int off = 128; off > 0; off >>= 1) {
        if (threadIdx.x < (unsigned)off) {
            smn[threadIdx.x] = fminf(smn[threadIdx.x], smn[threadIdx.x + off]);
            smx[threadIdx.x] = fmaxf(smx[threadIdx.x], smx[threadIdx.x + off]);
        }
        __syncthreads();
    }
    if (threadIdx.x == 0) {
        minmax[bc]      = smn[0];
        minmax[16 + bc] = smx[0];
    }
}

// ---------------------------------------------------------------------------
// Kernel 2: 1-NN in 2D for the 64x64 mesh samples; fused gather of the
// selected point's first 3 channels into sel[b][s][4].
// Packed-key argmin: exact lower-index tie-break, branchless (v_min_u64).
// ---------------------------------------------------------------------------
__global__ void knn1_kernel(const float* __restrict__ x,
                            const float* __restrict__ minmax,
                            float* __restrict__ sel) {
    int b = blockIdx.x >> 4;                        // 16 blocks per batch
    int s = ((blockIdx.x & 15) << 8) + threadIdx.x; // sample id in [0,4096)
    const float* xb = x + (size_t)b * 5 * NPTS;

    float mn0 = minmax[b * 2 + 0], mn1 = minmax[b * 2 + 1];
    float mx0 = minmax[16 + b * 2 + 0], mx1 = minmax[16 + b * 2 + 1];
    int i = s & 63, j = s >> 6;                     // m01 = mesh[s%64], m02 = mesh[s/64]
    float q0 = ((float)i * (1.0f / 64.0f)) * (mx0 - mn0) + mn0;
    float q1 = ((float)j * (1.0f / 64.0f)) * (mx1 - mn1) + mn1;

    __shared__ float l0[CHUNK], l1[CHUNK];
    unsigned long long best = ~0ull;
    for (int base = 0; base < NPTS; base += CHUNK) {
        __syncthreads();
        for (int t = threadIdx.x; t < CHUNK; t += 256) {
            l0[t] = xb[base + t];
            l1[t] = xb[NPTS + base + t];
        }
        __syncthreads();
        for (int t = 0; t < CHUNK; ++t) {
            float dx = l0[t] - q0;
            float dy = l1[t] - q1;
            float d = dx * dx + dy * dy;
            best = umin64(best, pack_key(d, base + t));
        }
    }
    int bidx = (int)(best & 0xFFFFFFFFull);
    float* sq = sel + ((size_t)b * NSAMP + s) * 4;
    sq[0] = xb[bidx];
    sq[1] = xb[NPTS + bidx];
    sq[2] = xb[2 * NPTS + bidx];
    sq[3] = 0.0f;
}

// ---------------------------------------------------------------------------
// Kernel 3: 16-NN in 3D via V_WMMA_F32_16X16X4_F32.
// Block = 256 threads = 8 waves; each wave owns 16 queries; block owns 128.
// Grid = 8 batches * 32 blocks.
//
// Fragment layouts per ISA 7.12.2 (32-bit A 16x4):
//   A: lanes 0-15 -> row M=lane,    VGPR0=K0(x),  VGPR1=K1(y)
//      lanes 16-31 -> row M=lane-16, VGPR0=K2(z),  VGPR1=K3(|p|^2)
//   B (4x16), mirrored: VGPR0: half0 = -2qx (K0), half1 = -2qz (K2)
//                        VGPR1: half0 = -2qy (K1), half1 = 1    (K3)
// ---------------------------------------------------------------------------
__global__ void knn2_kernel(const float* __restrict__ x,
                            const float* __restrict__ sel,
                            int* __restrict__ idx2) {
    int b = blockIdx.x >> 5;                 // 32 blocks per batch
    int qblk = (blockIdx.x & 31) * 128;      // query base within batch
    int lane = threadIdx.x & 31;
    int wave = threadIdx.x >> 5;
    int qn = lane & 15;                      // query column (N) of this lane
    int half = lane >> 4;                    // 0: points M=0..7, 1: M=8..15
    int q = qblk + wave * 16 + qn;           // query id within batch

    const float* xb = x + (size_t)b * 5 * NPTS;
    const float* sq = sel + ((size_t)b * NSAMP + q) * 4;
    float qx = sq[0], qy = sq[1], qz = sq[2];

    v2f bfrag;
    bfrag.x = half ? (-2.0f * qz) : (-2.0f * qx);
    bfrag.y = half ? 1.0f : (-2.0f * qy);

#if HAVE_ASYNC_LDS
    __shared__ __align__(16) float lx[2][CHUNK];
    __shared__ __align__(16) float ly[2][CHUNK];
    __shared__ __align__(16) float lz[2][CHUNK];
#else
    __shared__ __align__(16) float lx[1][CHUNK];
    __shared__ __align__(16) float ly[1][CHUNK];
    __shared__ __align__(16) float lz[1][CHUNK];
#endif
    __shared__ unsigned long long mergeK[8][16][32];

    unsigned long long keys[KNN];
#pragma unroll
    for (int i = 0; i < KNN; ++i) keys[i] = ~0ull;

#if HAVE_ASYNC_LDS
    // ---- async double-buffered chunk pipeline ----
    {
        int t4 = threadIdx.x << 2;           // 4 consecutive floats per lane
        __builtin_amdgcn_global_load_async_to_lds_b128(
            (gv4i*)(xb + t4),            (lv4i*)&lx[0][t4], 0, 0);
        __builtin_amdgcn_global_load_async_to_lds_b128(
            (gv4i*)(xb + NPTS + t4),     (lv4i*)&ly[0][t4], 0, 0);
        __builtin_amdgcn_global_load_async_to_lds_b128(
            (gv4i*)(xb + 2 * NPTS + t4), (lv4i*)&lz[0][t4], 0, 0);
        WAIT_ASYNC();
    }
    __syncthreads();
#endif

    for (int cc = 0; cc < NCHUNK; ++cc) {
        int cbase = cc * CHUNK;
#if HAVE_ASYNC_LDS
        int cur = cc & 1;
        if (cc + 1 < NCHUNK) {               // overlap next chunk's DMA
            int nb = cbase + CHUNK;
            int nxt = cur ^ 1;
            int t4 = threadIdx.x << 2;
            __builtin_amdgcn_global_load_async_to_lds_b128(
                (gv4i*)(xb + nb + t4),            (lv4i*)&lx[nxt][t4], 0, 0);
            __builtin_amdgcn_global_load_async_to_lds_b128(
                (gv4i*)(xb + NPTS + nb + t4),     (lv4i*)&ly[nxt][t4], 0, 0);
            __builtin_amdgcn_global_load_async_to_lds_b128(
                (gv4i*)(xb + 2 * NPTS + nb + t4), (lv4i*)&lz[nxt][t4], 0, 0);
        }
#else
        int cur = 0;
        __syncthreads();
        for (int t = threadIdx.x; t < CHUNK; t += 256) {
            lx[0][t] = xb[cbase + t];
            ly[0][t] = xb[NPTS + cbase + t];
            lz[0][t] = xb[2 * NPTS + cbase + t];
        }
        __syncthreads();
#endif
        const float* lxb = &lx[cur][0];
        const float* lyb = &ly[cur][0];
        const float* lzb = &lz[cur][0];

        // software pipeline: prefetch next tile's coords before using current
        float px = lxb[qn], py = lyb[qn], pz = lzb[qn];
        for (int t = 0; t < NTILE; ++t) {
            float cx = px, cy = py, cz = pz;
            int prn = (t + 1 < NTILE ? t + 1 : t) * 16 + qn;
            px = lxb[prn];
            py = lyb[prn];
            pz = lzb[prn];
            float pp = fmaf(cz, cz, fmaf(cy, cy, cx * cx));
            v2f afrag;
            afrag.x = half ? cz : cx;
            afrag.y = half ? pp : cy;
            v8f c = {};
            c = __builtin_amdgcn_wmma_f32_16x16x4_f32(
                    false, afrag, false, bfrag, (short)0, c, false, false);
            // c[r] = dist' of point (cbase + t*16 + r + 8*half) vs query qn
            float m0 = fminf(fminf(c[0], c[1]), fminf(c[2], c[3]));
            float m1 = fminf(fminf(c[4], c[5]), fminf(c[6], c[7]));
            unsigned cu = ord_f32(fminf(m0, m1));
            unsigned kth = (unsigned)(keys[KNN - 1] >> 32);
            // wave-uniform fast-skip (vcc branch, EXEC untouched)
            if (__builtin_amdgcn_ballot_w32(cu <= kth) != 0u) {
                int pbase = cbase + t * 16 + (half << 3);
#pragma unroll
                for (int r = 0; r < 8; ++r) {
                    unsigned long long key = pack_key(c[r], pbase + r);
                    // branchless sorted insertion (v_min_u64 / v_max_u64)
#pragma unroll
                    for (int i = KNN - 1; i >= 1; --i)
                        keys[i] = umax64(keys[i - 1], umin64(keys[i], key));
                    keys[0] = umin64(keys[0], key);
                }
            }
        }
#if HAVE_ASYNC_LDS
        WAIT_ASYNC();                        // own DMAs done...
        __syncthreads();                     // ...and everyone's, chunk retired
#endif
    }

    // merge the two half-lists per query (points 0..7 vs 8..15 interleave)
#pragma unroll
    for (int i = 0; i < KNN; ++i)
        mergeK[wave][qn][half * 16 + i] = keys[i];
    __syncthreads();
    if (half == 0) {
        int ia = 0, ib = 0;
        int* outp = idx2 + ((size_t)b * NSAMP + q) * KNN;
        for (int k = 0; k < KNN; ++k) {
            unsigned long long ka = mergeK[wave][qn][ia];
            unsigned long long kb = mergeK[wave][qn][16 + ib];
            bool takeA = ka < kb;            // key compare = (dist, idx) lexicographic
            outp[k] = (int)((takeA ? ka : kb) & 0xFFFFFFFFull);
            if (takeA) ++ia; else ++ib;
        }
    }
}

// ---------------------------------------------------------------------------
// Kernel 4: gather all 5 channels at idx2 -> out[b][c][q][k].
// ---------------------------------------------------------------------------
__global__ void gather_kernel(const float* __restrict__ x,
                              const int* __restrict__ idx2,
                              float* __restrict__ out) {
    int gid = blockIdx.x * 256 + threadIdx.x;   // b*65536 + q*16 + k
    int b = gid >> 16;
    int qk = gid & 65535;
    int idx = idx2[gid];
    const float* xb = x + (size_t)b * 5 * NPTS;
    float* ob = out + (size_t)b * 5 * NSAMP * KNN;
#pragma unroll
    for (int c = 0; c < 5; ++c)
        ob[(size_t)c * NSAMP * KNN + qk] = xb[(size_t)c * NPTS + idx];
}

// ---------------------------------------------------------------------------
// Launch. Workspace layout (floats):
//   [0, 64)    : minmax (16 min + 16 max)
//   [64, +128K): sel, 8*4096 float4
//   then       : idx2, 8*4096*16 int           (total ~2.6 MB)
// ---------------------------------------------------------------------------
extern "C" void kernel_launch(void* const* d_in, const int* in_sizes, int n_in,
                              void* d_out, int out_size, void* d_ws, size_t ws_size,
                              hipStream_t stream) {
    const float* x = (const float*)d_in[0];
    (void)in_sizes; (void)n_in; (void)out_size; (void)ws_size;

    float* ws = (float*)d_ws;
    float* minmax = ws;
    float* sel = ws + 64;
    int* idx2 = (int*)(ws + 64 + BATCHES * NSAMP * 4);
    float* out = (float*)d_out;

    minmax_kernel<<<BATCHES * 2, 256, 0, stream>>>(x, minmax);
    knn1_kernel<<<BATCHES * 16, 256, 0, stream>>>(x, minmax, sel);
    knn2_kernel<<<BATCHES * 32, 256, 0, stream>>>(x, sel, idx2);
    gather_kernel<<<BATCHES * NSAMP * KNN / 256, 256, 0, stream>>>(x, idx2, out);
}